// MetaGATGRU_54322746359884
// MI455X (gfx1250) — compile-verified
//
#include <hip/hip_runtime.h>
#include <cstddef>
#include <cstdint>

// ---------------- problem constants (from reference setup_inputs) ----------
#define B_    2
#define N_    5000
#define BN    10000          // B_*N_
#define T_    12
#define FIN   32
#define M_    32
#define H_    4
#define C_    16
#define HD    64             // H_*C_
#define O_    32
#define E_    80000
#define ETOT  85000          // E_ + N_ (self loops appended)
#define NEG_SLOPE 0.2f
#define XSST  1024           // xs row stride: 16 (padded T) * 64

// ---------------- workspace layout (float offsets) -------------------------
static const size_t OFF_WG   = 0;                         // [BN,32,64]      20,480,000
static const size_t OFF_WX   = 20480000;                  // [3,BN,64,64]   122,880,000
static const size_t OFF_WH   = 143360000;                 // [3,BN,64,64]   122,880,000
static const size_t OFF_BX   = 266240000;                 // [3,BN,64]        1,920,000
static const size_t OFF_BH   = 268160000;                 // [3,BN,64]        1,920,000
static const size_t OFF_XS   = 270080000;                 // [BN,16,64] pad  10,240,000
static const size_t OFF_ASRC = 280320000;                 // [BN,12,4]          480,000
static const size_t OFF_ADST = 280800000;                 // [BN,12,4]          480,000
static const size_t OFF_SEGM = 281280000;                 // [BN,4] (uint)       40,000
static const size_t OFF_DEN  = 281320000;                 // [BN,4]              40,000
static const size_t OFF_EX   = 281360000;                 // [ETOT,2,4]         680,000
static const size_t OFF_AGG  = 282040000;                 // [BN,64]            640,000
static const size_t OFF_H    = 282680000;                 // [BN,64]            640,000

// ---------------- vector types / WMMA helpers -------------------------------
typedef __attribute__((ext_vector_type(2))) float v2f;
typedef __attribute__((ext_vector_type(4))) float v4f;
typedef __attribute__((ext_vector_type(8))) float v8f;

static __device__ __forceinline__ v8f wmma_f32_4(v2f a, v2f b, v8f c) {
  // V_WMMA_F32_16X16X4_F32 : D(16x16) = A(16x4) * B(4x16) + C
  return __builtin_amdgcn_wmma_f32_16x16x4_f32(false, a, false, b, (short)0, c,
                                               false, false);
}

// async global->LDS copy (ASYNCcnt-tracked); per-lane 16B chunk
static __device__ __forceinline__ void async_ld_b128(unsigned lds_off,
                                                     const void* gptr) {
  asm volatile("global_load_async_to_lds_b128 %0, %1, off"
               :: "v"(lds_off), "v"((unsigned long long)(uintptr_t)gptr)
               : "memory");
}
static __device__ __forceinline__ void wait_async0() {
#if __has_builtin(__builtin_amdgcn_s_wait_asynccnt)
  __builtin_amdgcn_s_wait_asynccnt(0);
#else
  asm volatile("s_wait_asynccnt 0x0" ::: "memory");
#endif
}

// order-preserving float<->uint map for atomicMax on floats
static __device__ __forceinline__ unsigned fenc(float f) {
  unsigned u = __float_as_uint(f);
  return (u & 0x80000000u) ? ~u : (u | 0x80000000u);
}
static __device__ __forceinline__ float fdec(unsigned u) {
  unsigned v = (u & 0x80000000u) ? (u & 0x7FFFFFFFu) : ~u;
  return __uint_as_float(v);
}

// ---------------- generic shared-B WMMA GEMM --------------------------------
// D[BN x ncols] = A[BN x K] @ Bm[K x ncols] + bias[ncols]
// one 16-row tile per block; waves in block stripe the 16-col tiles.
template <int K>
__global__ void gemm32_wmma(const float* __restrict__ A,
                            const float* __restrict__ Bm,
                            const float* __restrict__ bias,
                            float* __restrict__ D, int ncols) {
  const int lane   = threadIdx.x & 31;
  const int wave   = threadIdx.x >> 5;
  const int nwaves = blockDim.x >> 5;
  const int rowTile = blockIdx.x;           // 16 rows per tile, BN % 16 == 0
  const int row16  = lane & 15;             // A row / C col within tile
  const int khalf  = (lane >> 4) * 2;       // K sub-pair selected by lane half
  const int mbase  = (lane >> 4) * 8;       // C/D row base per ISA layout
  constexpr int KF = K / 4;

  v2f aF[KF];
  const float* Arow = A + (size_t)(rowTile * 16 + row16) * K;
#pragma unroll
  for (int ks = 0; ks < KF; ++ks)
    aF[ks] = *(const v2f*)(Arow + ks * 4 + khalf);   // global_load_b64

  const int ctiles = ncols >> 4;
  for (int ct = wave; ct < ctiles; ct += nwaves) {
    const int col = ct * 16 + row16;
    // prefetch next B tile this wave will touch (global_prefetch_b8)
    if (ct + nwaves < ctiles)
      __builtin_prefetch(&Bm[(size_t)khalf * ncols + (ct + nwaves) * 16 + row16], 0, 1);
    v8f acc = {};
#pragma unroll
    for (int ks = 0; ks < KF; ++ks) {
      v2f bF;
      bF.x = Bm[(size_t)(ks * 4 + khalf) * ncols + col];
      bF.y = Bm[(size_t)(ks * 4 + khalf + 1) * ncols + col];
      acc = wmma_f32_4(aF[ks], bF, acc);
    }
    const float bv = bias[col];
#pragma unroll
    for (int v = 0; v < 8; ++v)
      D[(size_t)(rowTile * 16 + mbase + v) * ncols + col] = acc[v] + bv;
  }
}

// ---------------- per-node x-projection (all 12 timesteps at once) ----------
// xs[node,t,:] = x[node,t,:] @ wg[node]  (A = 12x32 padded to 16, B = 32x64)
// xs is padded to 16 rows per node -> unconditional epilogue stores.
__global__ void xproj_wmma(const float* __restrict__ x,   // [BN,12,32]
                           const float* __restrict__ wg,  // [BN,32,64]
                           float* __restrict__ xs) {      // [BN,16,64]
  const int lane = threadIdx.x & 31;
  const int wave = threadIdx.x >> 5;
  const int node = blockIdx.x * (blockDim.x >> 5) + wave;  // exact: BN waves
  const int row16 = lane & 15;
  const int khalf = (lane >> 4) * 2;
  const int mbase = (lane >> 4) * 8;

  v2f aF[8];
  const float* xr = x + (size_t)node * (T_ * FIN) + row16 * FIN;
#pragma unroll
  for (int ks = 0; ks < 8; ++ks) {
    v2f a = {0.f, 0.f};
    if (row16 < T_) a = *(const v2f*)(xr + ks * 4 + khalf);
    aF[ks] = a;
  }
  const float* wgn = wg + (size_t)node * (FIN * HD);
#pragma unroll
  for (int ct = 0; ct < 4; ++ct) {
    const int col = ct * 16 + row16;
    v8f acc = {};
#pragma unroll
    for (int ks = 0; ks < 8; ++ks) {
      v2f bF;
      bF.x = wgn[(ks * 4 + khalf) * HD + col];
      bF.y = wgn[(ks * 4 + khalf + 1) * HD + col];
      acc = wmma_f32_4(aF[ks], bF, acc);
    }
#pragma unroll
    for (int v = 0; v < 8; ++v)
      xs[(size_t)node * XSST + (mbase + v) * HD + col] = acc[v];
  }
}

// ---------------- attention logits per (node,t,h) --------------------------
__global__ void attn_scores(const float* __restrict__ xs,
                            const float* __restrict__ att_src,
                            const float* __restrict__ att_dst,
                            float* __restrict__ asrc, float* __restrict__ adst) {
  const int tid = blockIdx.x * blockDim.x + threadIdx.x;  // node*48 + t*4 + h
  if (tid >= BN * T_ * H_) return;
  const int h = tid & 3;
  const int t = (tid >> 2) % T_;
  const int node = tid / (T_ * H_);
  const float* vp = xs + (size_t)node * XSST + t * HD + h * C_;
  float s = 0.f, d = 0.f;
#pragma unroll
  for (int c4 = 0; c4 < 4; ++c4) {
    const v4f v = *(const v4f*)(vp + c4 * 4);
    const v4f as = *(const v4f*)(att_src + h * C_ + c4 * 4);
    const v4f ad = *(const v4f*)(att_dst + h * C_ + c4 * 4);
#pragma unroll
    for (int j = 0; j < 4; ++j) {
      s = fmaf(v[j], as[j], s);
      d = fmaf(v[j], ad[j], d);
    }
  }
  asrc[tid] = s;
  adst[tid] = d;
}

// ---------------- per-step scratch init ------------------------------------
__global__ void step_init(unsigned* __restrict__ segm, float* __restrict__ den,
                          float* __restrict__ agg) {
  const int tid = blockIdx.x * blockDim.x + threadIdx.x;
  if (tid < BN * H_) { segm[tid] = 0u; den[tid] = 0.f; }
  if (tid < BN * HD) agg[tid] = 0.f;
}

__global__ void zero_f32(float* __restrict__ p, int n) {
  const int tid = blockIdx.x * blockDim.x + threadIdx.x;
  if (tid < n) p[tid] = 0.f;
}

// ---------------- edge score helper ----------------------------------------
static __device__ __forceinline__ float edge_score(const float* asrc,
                                                   const float* adst, int b,
                                                   int s, int d, int t, int h) {
  float a = asrc[(size_t)(b * N_ + s) * (T_ * H_) + t * H_ + h] +
            adst[(size_t)(b * N_ + d) * (T_ * H_) + t * H_ + h];
  return (a >= 0.f) ? a : NEG_SLOPE * a;
}

// pass 1: segment max via encoded atomicMax
__global__ void edge_pass1(const int* __restrict__ ei,
                           const float* __restrict__ asrc,
                           const float* __restrict__ adst,
                           unsigned* __restrict__ segm, int t) {
  const int tid = blockIdx.x * blockDim.x + threadIdx.x;  // e*8 + b*4 + h
  if (tid >= ETOT * B_ * H_) return;
  const int e = tid >> 3, b = (tid >> 2) & 1, h = tid & 3;
  const int s = (e < E_) ? ei[e] : (e - E_);
  const int d = (e < E_) ? ei[E_ + e] : (e - E_);
  const float a = edge_score(asrc, adst, b, s, d, t, h);
  atomicMax(&segm[(b * N_ + d) * H_ + h], fenc(a));
}

// pass 2: exp(a - max), segment-sum denominator
__global__ void edge_pass2(const int* __restrict__ ei,
                           const float* __restrict__ asrc,
                           const float* __restrict__ adst,
                           const unsigned* __restrict__ segm,
                           float* __restrict__ den, float* __restrict__ exb,
                           int t) {
  const int tid = blockIdx.x * blockDim.x + threadIdx.x;
  if (tid >= ETOT * B_ * H_) return;
  const int e = tid >> 3, b = (tid >> 2) & 1, h = tid & 3;
  const int s = (e < E_) ? ei[e] : (e - E_);
  const int d = (e < E_) ? ei[E_ + e] : (e - E_);
  const float a = edge_score(asrc, adst, b, s, d, t, h);
  const float ex = __expf(a - fdec(segm[(b * N_ + d) * H_ + h]));
  exb[tid] = ex;
  atomicAdd(&den[(b * N_ + d) * H_ + h], ex);
}

// pass 3: alpha-weighted scatter-add of messages
__global__ void edge_pass3(const int* __restrict__ ei,
                           const float* __restrict__ xs,
                           const float* __restrict__ den,
                           const float* __restrict__ exb,
                           float* __restrict__ agg, int t) {
  const int tid = blockIdx.x * blockDim.x + threadIdx.x;
  if (tid >= ETOT * B_ * H_) return;
  const int e = tid >> 3, b = (tid >> 2) & 1, h = tid & 3;
  const int s = (e < E_) ? ei[e] : (e - E_);
  const int d = (e < E_) ? ei[E_ + e] : (e - E_);
  const float alpha = exb[tid] / (den[(b * N_ + d) * H_ + h] + 1e-16f);
  const float* msg = xs + (size_t)(b * N_ + s) * XSST + t * HD + h * C_;
  float* dstp = agg + (size_t)(b * N_ + d) * HD + h * C_;
#pragma unroll
  for (int c4 = 0; c4 < 4; ++c4) {
    const v4f m = *(const v4f*)(msg + c4 * 4);   // global_load_b128
#pragma unroll
    for (int j = 0; j < 4; ++j) atomicAdd(&dstp[c4 * 4 + j], m[j] * alpha);
  }
}

// ---------------- per-node meta-GRU gate application -----------------------
// 16 nodes / block (256 thr); m,h staged into LDS via async global->LDS.
// 16 threads per node, 4 outputs each -> all Wx/Wh traffic is b128, coalesced.
__global__ void __launch_bounds__(256) gru_step(
    const float* __restrict__ agg, const float* __restrict__ gat_bias,
    const float* __restrict__ Wx, const float* __restrict__ bx,
    const float* __restrict__ Wh, const float* __restrict__ bh,
    float* __restrict__ h) {
  __shared__ float sM[16 * HD];
  __shared__ float sH[16 * HD];
  const int tid = threadIdx.x;
  const int nodeBase = blockIdx.x * 16;

  // async stage: 2 x 4KB (256 chunks of 16B each), one chunk per lane per op
  {
    const int nd = tid >> 4, part = tid & 15;
    async_ld_b128((unsigned)(uintptr_t)&sM[nd * HD + part * 4],
                  agg + (size_t)(nodeBase + nd) * HD + part * 4);
    async_ld_b128((unsigned)(uintptr_t)&sH[nd * HD + part * 4],
                  h + (size_t)(nodeBase + nd) * HD + part * 4);
    wait_async0();
  }
  __syncthreads();
  // fold gat_bias into message
  for (int k = tid; k < 16 * HD; k += 256) sM[k] += gat_bias[k & (HD - 1)];
  __syncthreads();

  const int tn = tid >> 4;              // node within block
  const int o4 = (tid & 15) * 4;        // 4 outputs per thread
  const int node = nodeBase + tn;
  const float* smp = &sM[tn * HD];
  const float* shp = &sH[tn * HD];

  v4f gx[3], gh[3];
#pragma unroll
  for (int g = 0; g < 3; ++g) {
    const float* wx = Wx + ((size_t)g * BN + node) * (HD * HD) + o4;
    const float* wh = Wh + ((size_t)g * BN + node) * (HD * HD) + o4;
    v4f ax = *(const v4f*)(bx + ((size_t)g * BN + node) * HD + o4);
    v4f ah = *(const v4f*)(bh + ((size_t)g * BN + node) * HD + o4);
    for (int i = 0; i < HD; ++i) {
      ax += *(const v4f*)(wx + i * HD) * smp[i];   // global_load_b128
      ah += *(const v4f*)(wh + i * HD) * shp[i];
    }
    gx[g] = ax;
    gh[g] = ah;
  }
  const v4f hv = *(const v4f*)(shp + o4);
  v4f hn;
#pragma unroll
  for (int j = 0; j < 4; ++j) {
    const float r = 1.f / (1.f + __expf(-(gx[0][j] + gh[0][j])));
    const float z = 1.f / (1.f + __expf(-(gx[1][j] + gh[1][j])));
    const float c = tanhf(gx[2][j] + r * gh[2][j]);
    hn[j] = (1.f - z) * c + z * hv[j];
  }
  *(v4f*)(h + (size_t)node * HD + o4) = hn;       // global_store_b128
}

// ---------------- driver ----------------------------------------------------
extern "C" void kernel_launch(void* const* d_in, const int* in_sizes, int n_in,
                              void* d_out, int out_size, void* d_ws,
                              size_t ws_size, hipStream_t stream) {
  (void)in_sizes; (void)n_in; (void)out_size; (void)ws_size;
  const float* meta     = (const float*)d_in[0];
  const float* x        = (const float*)d_in[1];
  const int*   ei       = (const int*)d_in[2];
  const float* gat_wW   = (const float*)d_in[3];
  const float* gat_wb   = (const float*)d_in[4];
  const float* att_src  = (const float*)d_in[5];
  const float* att_dst  = (const float*)d_in[6];
  const float* gat_bias = (const float*)d_in[7];
  const float* gxwW     = (const float*)d_in[8];
  const float* gxwB     = (const float*)d_in[9];
  const float* gxbW     = (const float*)d_in[10];
  const float* gxbB     = (const float*)d_in[11];
  const float* ghwW     = (const float*)d_in[12];
  const float* ghwB     = (const float*)d_in[13];
  const float* ghbW     = (const float*)d_in[14];
  const float* ghbB     = (const float*)d_in[15];
  const float* out_W    = (const float*)d_in[16];
  const float* out_b    = (const float*)d_in[17];
  float* out = (float*)d_out;

  float* ws = (float*)d_ws;
  float* wg   = ws + OFF_WG;
  float* Wx   = ws + OFF_WX;
  float* Wh   = ws + OFF_WH;
  float* bx   = ws + OFF_BX;
  float* bh   = ws + OFF_BH;
  float* xs   = ws + OFF_XS;
  float* asrc = ws + OFF_ASRC;
  float* adst = ws + OFF_ADST;
  unsigned* segm = (unsigned*)(ws + OFF_SEGM);
  float* den  = ws + OFF_DEN;
  float* exb  = ws + OFF_EX;
  float* agg  = ws + OFF_AGG;
  float* hbuf = ws + OFF_H;

  const dim3 rowGrid(BN / 16);  // 625 sixteen-row tiles

  // ---- hypernetwork (time-invariant): WMMA GEMMs, shared B matrices ----
  gemm32_wmma<32><<<rowGrid, 128, 0, stream>>>(meta, gat_wW, gat_wb, wg, FIN * HD);
  for (int g = 0; g < 3; ++g) {
    gemm32_wmma<32><<<rowGrid, 256, 0, stream>>>(
        meta, gxwW + (size_t)g * M_ * HD * HD, gxwB + (size_t)g * HD * HD,
        Wx + (size_t)g * BN * HD * HD, HD * HD);
    gemm32_wmma<32><<<rowGrid, 256, 0, stream>>>(
        meta, ghwW + (size_t)g * M_ * HD * HD, ghwB + (size_t)g * HD * HD,
        Wh + (size_t)g * BN * HD * HD, HD * HD);
    gemm32_wmma<32><<<rowGrid, 128, 0, stream>>>(
        meta, gxbW + (size_t)g * M_ * HD, gxbB + (size_t)g * HD,
        bx + (size_t)g * BN * HD, HD);
    gemm32_wmma<32><<<rowGrid, 128, 0, stream>>>(
        meta, ghbW + (size_t)g * M_ * HD, ghbB + (size_t)g * HD,
        bh + (size_t)g * BN * HD, HD);
  }

  // ---- per-node x-projection for all 12 timesteps (1 wave == 1 node) ----
  xproj_wmma<<<dim3(BN / 4), 128, 0, stream>>>(x, wg, xs);
  attn_scores<<<dim3((BN * T_ * H_ + 255) / 256), 256, 0, stream>>>(
      xs, att_src, att_dst, asrc, adst);

  // ---- recurrent loop ----
  zero_f32<<<dim3((BN * HD + 255) / 256), 256, 0, stream>>>(hbuf, BN * HD);
  const int edgeThreads = ETOT * B_ * H_;
  const dim3 edgeGrid((edgeThreads + 255) / 256);
  for (int t = 0; t < T_; ++t) {
    step_init<<<dim3((BN * HD + 255) / 256), 256, 0, stream>>>(segm, den, agg);
    edge_pass1<<<edgeGrid, 256, 0, stream>>>(ei, asrc, adst, segm, t);
    edge_pass2<<<edgeGrid, 256, 0, stream>>>(ei, asrc, adst, segm, den, exb, t);
    edge_pass3<<<edgeGrid, 256, 0, stream>>>(ei, xs, den, exb, agg, t);
    gru_step<<<dim3(BN / 16), 256, 0, stream>>>(agg, gat_bias, Wx, bx, Wh, bh,
                                                hbuf);
    // out[t] = h @ out_W + out_b  (shared-B WMMA GEMM, K=64)
    gemm32_wmma<64><<<rowGrid, 64, 0, stream>>>(hbuf, out_W, out_b,
                                                out + (size_t)t * BN * O_, O_);
  }
}